// PinSageConv_24481313587344
// MI455X (gfx1250) — compile-verified
//
#include <hip/hip_runtime.h>

// MI455X / gfx1250 fused PinSageConv.
// f32 WMMA (V_WMMA_F32_16X16X4_F32): kernel is gather-bandwidth-bound
// (~512MB HBM vs 69 GFLOP), so f32 matrix math is free and exact.
// Gather staging uses GLOBAL_LOAD_ASYNC_TO_LDS_B128 (ASYNCcnt).
// LDS neighbor tile is stored (t, m, k)-major so that all 8 m-row fragments a
// lane needs sit within one 16-bit DS immediate window of a single base VGPR.

typedef __attribute__((ext_vector_type(2))) float v2f;
typedef __attribute__((ext_vector_type(8))) float v8f;

// exact param types per hipcc diagnostic: <4 x i32> with addrspace(1)/(3)
typedef int v4i_ __attribute__((vector_size(4 * sizeof(int))));
typedef __attribute__((address_space(1))) v4i_* as1_v4i;
typedef __attribute__((address_space(3))) v4i_* as3_v4i;

#if __has_builtin(__builtin_amdgcn_global_load_async_to_lds_b128)
#define HAVE_ASYNC_LDS 1
#else
#define HAVE_ASYNC_LDS 0
#endif

#define D 256
#define H 256
#define O 256
#define T 16
#define NPB 8                     // nodes per block
#define NB_ROWS (NPB * T)         // 128 neighbor rows staged per block
#define NB_K 260                  // row stride (256 + 4 pad)
#define NB_G (NPB * NB_K + 4)     // 2084: per-t group stride; 2084%64=36 ->
                                  // 36*l mod 64 = 16 distinct banks, b64-safe
#define CAT_W (D + H)             // 512
#define CAT_STRIDE 516            // 512 + 4 pad (516%64==4 -> conflict-free)

__device__ __forceinline__ float lrelu(float x) { return x > 0.f ? x : 0.01f * x; }

__device__ __forceinline__ v8f wmma_f32(v2f a, v2f b, v8f c) {
    // (neg_a, A, neg_b, B, c_mod, C, reuse_a, reuse_b)
    return __builtin_amdgcn_wmma_f32_16x16x4_f32(false, a, false, b, (short)0, c,
                                                 false, false);
}

__device__ __forceinline__ void async_copy16(const float* gsrc, float* ldst) {
#if HAVE_ASYNC_LDS
    __builtin_amdgcn_global_load_async_to_lds_b128((as1_v4i)gsrc, (as3_v4i)ldst,
                                                   0, 0);
#else
    *(float4*)ldst = *(const float4*)gsrc;
#endif
}

__device__ __forceinline__ void async_wait0() {
#if HAVE_ASYNC_LDS
#if __has_builtin(__builtin_amdgcn_s_wait_asynccnt)
    __builtin_amdgcn_s_wait_asynccnt(0);
#else
    asm volatile("s_wait_asynccnt 0x0" ::: "memory");
#endif
#endif
}

__launch_bounds__(256)
__global__ void pinsage_fused(const float* __restrict__ h,
                              const int*   __restrict__ nodeset,
                              const int*   __restrict__ nb_nodes,
                              const float* __restrict__ nb_w,
                              const float* __restrict__ Qw,
                              const float* __restrict__ Qb,
                              const float* __restrict__ Ww,
                              const float* __restrict__ Wb,
                              float* __restrict__ out)
{
    // ~130 KB neighbor tile; concat (16x516) + hnew (8x260) overlay it later.
    __shared__ __align__(16) float s_nb[16 * NB_G];
    __shared__ float s_w[NB_ROWS];
    __shared__ float s_winv[NPB];

    float* s_cat  = s_nb;                       // 16 rows x CAT_STRIDE
    float* s_hnew = s_nb + 16 * CAT_STRIDE;     // 8 rows x NB_K

    const int tid  = threadIdx.x;
    const int wv   = tid >> 5;       // wave 0..7 (wave32)
    const int lane = tid & 31;
    const int l16  = lane & 15;
    const int hi   = lane >> 4;      // which half of the wave
    const int blk  = blockIdx.x;
    const long long rowbase = (long long)blk * NB_ROWS;

    // ---------------- stage 128 gathered neighbor rows into LDS -------------
    // Logical row r = (node m = r>>4, neighbor t = r&15) stored at
    // s_nb[t*NB_G + m*NB_K + k] so GEMM1 fragments use immediate DS offsets.
    {
        const int sub = tid & 63;    // 64 lanes x 16B = one 1KB row
        const int rr  = tid >> 6;    // 4 rows per pass
        #pragma unroll 4
        for (int r0 = 0; r0 < NB_ROWS; r0 += 4) {
            const int r = r0 + rr;
            const long long src = (long long)nb_nodes[rowbase + r] * D;
            float* dst = &s_nb[(r & 15) * NB_G + (r >> 4) * NB_K + sub * 4];
            async_copy16(h + src + sub * 4, dst);
        }
    }
    if (tid < NB_ROWS) s_w[tid] = nb_w[rowbase + tid];
    if (tid < NPB) {
        float s = 0.f;
        for (int t = 0; t < T; ++t) s += nb_w[rowbase + tid * T + t];
        s_winv[tid] = (s == 0.f) ? 1.f : (1.f / s);   // safediv
    }
    async_wait0();
    __syncthreads();

    // ---------------- GEMM1: Y = nb @ Qw^T, then weighted-mean over T -------
    // Wave wv owns output-column tiles {wv, wv+8}; all 8 node m-tiles.
    const float* abase = &s_nb[l16 * NB_G + hi * 2];  // single base per lane

    float aggv[2][NPB];
    for (int ni = 0; ni < 2; ++ni) {
        const int h0 = (wv + ni * 8) * 16;
        const float* bbase = Qw + (h0 + l16) * D + hi * 2;  // B[k][n]=Qw[n][k]
        const float  qb    = Qb[h0 + l16];
        v8f acc[NPB];
        #pragma unroll
        for (int m = 0; m < NPB; ++m)
            #pragma unroll
            for (int r = 0; r < 8; ++r) acc[m][r] = 0.f;

        #pragma unroll 8
        for (int k0 = 0; k0 < D; k0 += 4) {
            const v2f b = *(const v2f*)(bbase + k0);            // L2-hot
            #pragma unroll
            for (int m = 0; m < NPB; ++m) {
                const v2f a = *(const v2f*)(abase + m * NB_K + k0);
                acc[m] = wmma_f32(a, b, acc[m]);
            }
        }
        // epilogue: +Qb, leaky_relu, weight, reduce 16 rows -> h_agg
        #pragma unroll
        for (int m = 0; m < NPB; ++m) {
            float s = 0.f;
            #pragma unroll
            for (int r = 0; r < 8; ++r) {        // D row M = r + 8*hi
                const float v = lrelu(acc[m][r] + qb);
                s += s_w[m * T + r + 8 * hi] * v;
            }
            s += __shfl_xor(s, 16, 32);          // combine the two half-waves
            aggv[ni][m] = s * s_winv[m];
        }
    }
    __syncthreads();   // everyone done reading s_nb; safe to overlay

    // ---------------- build concat = [h[nodeset] | h_agg] in LDS ------------
    {
        const int sub = tid & 63;
        const int rr  = tid >> 6;
        #pragma unroll
        for (int r0 = 0; r0 < NPB; r0 += 4) {
            const int r = r0 + rr;
            const long long src = (long long)nodeset[blk * NPB + r] * D;
            async_copy16(h + src + sub * 4, &s_cat[r * CAT_STRIDE + sub * 4]);
        }
        for (int i = tid; i < 8 * CAT_STRIDE; i += 256)   // zero pad rows 8..15
            s_cat[8 * CAT_STRIDE + i] = 0.f;
    }
    if (lane < 16) {
        for (int ni = 0; ni < 2; ++ni) {
            const int h0 = (wv + ni * 8) * 16;
            for (int m = 0; m < NPB; ++m)
                s_cat[m * CAT_STRIDE + D + h0 + l16] = aggv[ni][m];
        }
    }
    async_wait0();
    __syncthreads();

    // ---------------- GEMM2: h_new = lrelu(concat @ Ww^T + Wb) --------------
    const float* a2ptr = &s_cat[l16 * CAT_STRIDE + hi * 2];
    float hval[2][8];
    for (int ni = 0; ni < 2; ++ni) {
        const int o0 = (wv + ni * 8) * 16;
        const float* bbase = Ww + (o0 + l16) * CAT_W + hi * 2;
        const float  wb    = Wb[o0 + l16];
        v8f acc;
        #pragma unroll
        for (int r = 0; r < 8; ++r) acc[r] = 0.f;

        #pragma unroll 8
        for (int k0 = 0; k0 < CAT_W; k0 += 4) {
            const v2f b = *(const v2f*)(bbase + k0);
            const v2f a = *(const v2f*)(a2ptr + k0);
            acc = wmma_f32(a, b, acc);
        }
        #pragma unroll
        for (int r = 0; r < 8; ++r) hval[ni][r] = lrelu(acc[r] + wb);
    }
    // rows 0..7 of the D tile (lanes 0..15, hi==0) are the valid 8 nodes
    if (lane < 16) {
        for (int ni = 0; ni < 2; ++ni) {
            const int o0 = (wv + ni * 8) * 16;
            for (int r = 0; r < 8; ++r)
                s_hnew[r * NB_K + o0 + l16] = hval[ni][r];
        }
    }
    __syncthreads();

    // ---------------- per-row L2 normalize + store (wave wv -> node wv) -----
    {
        float v[8];
        float ss = 0.f;
        #pragma unroll
        for (int j = 0; j < 8; ++j) {
            v[j] = s_hnew[wv * NB_K + lane + 32 * j];
            ss += v[j] * v[j];
        }
        #pragma unroll
        for (int off = 16; off > 0; off >>= 1) ss += __shfl_xor(ss, off, 32);
        const float nrm = sqrtf(ss);
        const float inv = (nrm == 0.f) ? 1.f : (1.f / nrm);   // safediv
        float* orow = out + ((long long)blk * NPB + wv) * O;
        #pragma unroll
        for (int j = 0; j < 8; ++j) orow[lane + 32 * j] = v[j] * inv;
    }
}

extern "C" void kernel_launch(void* const* d_in, const int* in_sizes, int n_in,
                              void* d_out, int out_size, void* d_ws, size_t ws_size,
                              hipStream_t stream) {
    const float* h       = (const float*)d_in[0];
    const int*   nodeset = (const int*)d_in[1];     // harness delivers int32 indices
    const int*   nb_node = (const int*)d_in[2];
    const float* nb_w    = (const float*)d_in[3];
    const float* Qw      = (const float*)d_in[4];
    const float* Qb      = (const float*)d_in[5];
    const float* Ww      = (const float*)d_in[6];
    const float* Wb      = (const float*)d_in[7];
    float* out = (float*)d_out;

    const int N = in_sizes[1];          // 32768 nodes
    const int blocks = N / NPB;         // 4096
    pinsage_fused<<<blocks, 256, 0, stream>>>(h, nodeset, nb_node, nb_w,
                                              Qw, Qb, Ww, Wb, out);
}